// VoxelBracketPredictor_v2_11579231830189
// MI455X (gfx1250) — compile-verified
//
#include <hip/hip_runtime.h>
#include <hip/hip_bf16.h>

// ---------------------------------------------------------------------------
// VoxelBracketPredictor for MI455X (gfx1250, wave32, WMMA).
//
// Cost model: pooling reads feat = 1M x 96 x 4B = 384 MB  (~16.5 us @ 23.3 TB/s).
// All GEMMs total ~1.2 GFLOP on [4096 x *] activations -> negligible vs pooling.
// => stream pooling at full BW; run MLPs with native f32 WMMA (16x16x4).
// ---------------------------------------------------------------------------

typedef __attribute__((ext_vector_type(2))) float v2f;
typedef __attribute__((ext_vector_type(8))) float v8f;

#define NPTS 1048576
#define BSEG 4096
#define CCH  96

// ---------------------------------------------------------------------------
// 1) Segment mean pooling. segment_ids are sorted, so each block binary-
//    searches its [start,end) range. blockDim.x == 96, one channel per thread.
//    Fully deterministic (no atomics), coalesced 384B row reads.
// ---------------------------------------------------------------------------
__global__ __launch_bounds__(96) void pool_kernel(
    const float* __restrict__ feat, const int* __restrict__ seg,
    float* __restrict__ pooled) {
  const int b = blockIdx.x;
  const int c = threadIdx.x;

  // lower_bound(seg, b) and lower_bound(seg, b+1); redundant per-thread search
  // (20 iters) is cheaper than an LDS broadcast dance.
  int lo = 0, hi = NPTS;
  while (lo < hi) { int mid = (lo + hi) >> 1; if (seg[mid] < b) lo = mid + 1; else hi = mid; }
  const int start = lo;
  hi = NPTS;
  while (lo < hi) { int mid = (lo + hi) >> 1; if (seg[mid] < b + 1) lo = mid + 1; else hi = mid; }
  const int end = lo;

  float s = 0.f;
  for (int i = start; i < end; ++i) s += feat[(long)i * CCH + c];
  const float cnt = (float)(end - start);
  pooled[(long)b * CCH + c] = s / fmaxf(cnt, 1.0f);
}

// ---------------------------------------------------------------------------
// 2) Batched GEMM: C[z] = A[z][M,K] @ W[z][K,N] + bias[z][N]   (row-major)
//    One 16x16 output tile per wave, 4 waves per block.
//    grid = (N/64, M/16, nbatch). K % 4 == 0, N % 64 == 0, M % 16 == 0.
//
//    V_WMMA_F32_16X16X4_F32 fragment layout (ISA 7.12.2):
//      A 16x4 : lanes 0-15 -> M=lane, VGPR0=K0, VGPR1=K1
//               lanes16-31 -> M=lane-16, VGPR0=K2, VGPR1=K3
//      B 4x16 : rows striped across lanes within a VGPR:
//               VGPR0: lanes0-15 = row K0, lanes16-31 = row K2
//               VGPR1: lanes0-15 = row K1, lanes16-31 = row K3
//      C/D    : VGPR r: lanes0-15 -> row r, col=lane; lanes16-31 -> row r+8
// ---------------------------------------------------------------------------
__global__ __launch_bounds__(128) void wmma_gemm_f32(
    const float* __restrict__ A, const float* __restrict__ W,
    const float* __restrict__ bias, float* __restrict__ Cout,
    int K, int N,
    long A_bstride, long W_bstride, long b_bstride, long C_bstride) {
  const int lane = threadIdx.x & 31;
  const int wave = threadIdx.x >> 5;
  const int half = lane >> 4;   // 0: lanes 0-15, 1: lanes 16-31
  const int l    = lane & 15;

  const int n0 = (blockIdx.x * 4 + wave) * 16;
  const int m0 = blockIdx.y * 16;
  const long z = blockIdx.z;

  A    += z * A_bstride;
  W    += z * W_bstride;
  bias += z * b_bstride;
  Cout += z * C_bstride;

  v8f acc = {};
  for (int kk = 0; kk < K; kk += 4) {
    // A fragment
    const float* arow = A + (long)(m0 + l) * K + kk + half * 2;
    v2f a; a.x = arow[0]; a.y = arow[1];
    // B fragment
    const float* wrow = W + (long)(kk + half * 2) * N + n0 + l;
    v2f bfrag; bfrag.x = wrow[0]; bfrag.y = wrow[N];
    acc = __builtin_amdgcn_wmma_f32_16x16x4_f32(
        /*neg_a=*/false, a, /*neg_b=*/false, bfrag,
        /*c_mod=*/(short)0, acc, /*reuse_a=*/false, /*reuse_b=*/false);
  }

  const float bv = bias[n0 + l];
#pragma unroll
  for (int r = 0; r < 8; ++r) {
    const int row = m0 + half * 8 + r;
    Cout[(long)row * N + n0 + l] = acc[r] + bv;
  }
}

// ---------------------------------------------------------------------------
// 3) In-place LayerNorm + ReLU, one wave32 per row. gamma/beta may be
//    per-head: head index = row / head_rows.
// ---------------------------------------------------------------------------
__global__ __launch_bounds__(32) void ln_relu_kernel(
    float* __restrict__ X, const float* __restrict__ gamma,
    const float* __restrict__ beta, int N, int head_rows) {
  const int row  = blockIdx.x;
  const int lane = threadIdx.x;
  const int hoff = (row / head_rows) * N;
  float* x = X + (long)row * N;

  float s = 0.f, sq = 0.f;
  for (int i = lane; i < N; i += 32) { float v = x[i]; s += v; sq += v * v; }
#pragma unroll
  for (int off = 16; off; off >>= 1) {
    s  += __shfl_xor(s,  off, 32);
    sq += __shfl_xor(sq, off, 32);
  }
  const float invn = 1.0f / (float)N;
  const float mean = s * invn;
  const float var  = sq * invn - mean * mean;
  const float rstd = rsqrtf(var + 1e-5f);
  for (int i = lane; i < N; i += 32) {
    float v = (x[i] - mean) * rstd * gamma[hoff + i] + beta[hoff + i];
    x[i] = fmaxf(v, 0.0f);
  }
}

// ---------------------------------------------------------------------------
// 4) Fused tiny output layers + class routing:
//    coarse = h2c[b,:128] @ cW3[128,3] + cb3
//    off    = r2[k,b,:64] @ rW3[k][64,3] + rb3[k],  k = class_ids[b]
//    out[b] = coarse + off
// ---------------------------------------------------------------------------
__global__ __launch_bounds__(256) void final_kernel(
    const float* __restrict__ h2c, const float* __restrict__ r2,
    const int* __restrict__ cls,
    const float* __restrict__ cW3, const float* __restrict__ cb3,
    const float* __restrict__ rW3, const float* __restrict__ rb3,
    float* __restrict__ out) {
  const int b = blockIdx.x * blockDim.x + threadIdx.x;
  if (b >= BSEG) return;

  float o0 = cb3[0], o1 = cb3[1], o2 = cb3[2];
  const float* h = h2c + (long)b * 128;
#pragma unroll 4
  for (int i = 0; i < 128; ++i) {
    const float v = h[i];
    o0 += v * cW3[i * 3 + 0];
    o1 += v * cW3[i * 3 + 1];
    o2 += v * cW3[i * 3 + 2];
  }

  const int k = cls[b];
  const float* r = r2 + ((long)k * BSEG + b) * 64;
  const float* w = rW3 + (long)k * 64 * 3;
  float p0 = rb3[k * 3 + 0], p1 = rb3[k * 3 + 1], p2 = rb3[k * 3 + 2];
#pragma unroll 4
  for (int i = 0; i < 64; ++i) {
    const float v = r[i];
    p0 += v * w[i * 3 + 0];
    p1 += v * w[i * 3 + 1];
    p2 += v * w[i * 3 + 2];
  }

  out[b * 3 + 0] = o0 + p0;
  out[b * 3 + 1] = o1 + p1;
  out[b * 3 + 2] = o2 + p2;
}

// ---------------------------------------------------------------------------
// Launch
// ---------------------------------------------------------------------------
extern "C" void kernel_launch(void* const* d_in, const int* in_sizes, int n_in,
                              void* d_out, int out_size, void* d_ws, size_t ws_size,
                              hipStream_t stream) {
  const float* feat = (const float*)d_in[0];
  const int*   seg  = (const int*)  d_in[1];
  const int*   cls  = (const int*)  d_in[2];
  const float* cW1  = (const float*)d_in[3];
  const float* cb1  = (const float*)d_in[4];
  const float* cg1  = (const float*)d_in[5];
  const float* cbe1 = (const float*)d_in[6];
  const float* cW2  = (const float*)d_in[7];
  const float* cb2  = (const float*)d_in[8];
  const float* cg2  = (const float*)d_in[9];
  const float* cbe2 = (const float*)d_in[10];
  const float* cW3  = (const float*)d_in[11];
  const float* cb3  = (const float*)d_in[12];
  const float* rW1  = (const float*)d_in[13];
  const float* rb1  = (const float*)d_in[14];
  const float* rg1  = (const float*)d_in[15];
  const float* rbe1 = (const float*)d_in[16];
  const float* rW2  = (const float*)d_in[17];
  const float* rb2  = (const float*)d_in[18];
  const float* rg2  = (const float*)d_in[19];
  const float* rbe2 = (const float*)d_in[20];
  const float* rW3  = (const float*)d_in[21];
  const float* rb3  = (const float*)d_in[22];

  float* ws     = (float*)d_ws;
  float* pooled = ws;                        // [4096, 96]
  float* h1c    = pooled + (long)BSEG * 96;  // [4096, 256]
  float* h2c    = h1c + (long)BSEG * 256;    // [4096, 128]
  float* r1     = h2c + (long)BSEG * 128;    // [4, 4096, 128]
  float* r2     = r1 + 4L * BSEG * 128;      // [4, 4096, 64]

  // 1) pooling (bandwidth-bound pass over 384 MB)
  pool_kernel<<<BSEG, 96, 0, stream>>>(feat, seg, pooled);

  // 2) coarse head
  //    L1: [4096,96] @ [96,256]
  wmma_gemm_f32<<<dim3(256 / 64, BSEG / 16, 1), 128, 0, stream>>>(
      pooled, cW1, cb1, h1c, 96, 256, 0, 0, 0, 0);
  ln_relu_kernel<<<BSEG, 32, 0, stream>>>(h1c, cg1, cbe1, 256, 1 << 30);
  //    L2: [4096,256] @ [256,128]
  wmma_gemm_f32<<<dim3(128 / 64, BSEG / 16, 1), 128, 0, stream>>>(
      h1c, cW2, cb2, h2c, 256, 128, 0, 0, 0, 0);
  ln_relu_kernel<<<BSEG, 32, 0, stream>>>(h2c, cg2, cbe2, 128, 1 << 30);

  // 3) refinement heads, batched over z = 4 heads
  //    L1: [4096,96] @ [96,128] x4 (A shared across heads -> stride 0)
  wmma_gemm_f32<<<dim3(128 / 64, BSEG / 16, 4), 128, 0, stream>>>(
      pooled, rW1, rb1, r1, 96, 128,
      /*A*/ 0, /*W*/ 96L * 128, /*b*/ 128, /*C*/ (long)BSEG * 128);
  ln_relu_kernel<<<4 * BSEG, 32, 0, stream>>>(r1, rg1, rbe1, 128, BSEG);
  //    L2: [4096,128] @ [128,64] x4
  wmma_gemm_f32<<<dim3(64 / 64, BSEG / 16, 4), 128, 0, stream>>>(
      r1, rW2, rb2, r2, 128, 64,
      /*A*/ (long)BSEG * 128, /*W*/ 128L * 64, /*b*/ 64, /*C*/ (long)BSEG * 64);
  ln_relu_kernel<<<4 * BSEG, 32, 0, stream>>>(r2, rg2, rbe2, 64, BSEG);

  // 4) tiny output layers + routing + add
  final_kernel<<<(BSEG + 255) / 256, 256, 0, stream>>>(
      h2c, r2, cls, cW3, cb3, rW3, rb3, (float*)d_out);
}